// GCN_85349590106379
// MI455X (gfx1250) — compile-verified
//
#include <hip/hip_runtime.h>
#include <hip/hip_bf16.h>

typedef float v2f __attribute__((ext_vector_type(2)));
typedef float v8f __attribute__((ext_vector_type(8)));

#define CH       128
#define N_GRAPHS 1024

// ---------------------------------------------------------------------------
// degree / normalization
// ---------------------------------------------------------------------------
__global__ void k_init_deg(float* __restrict__ deg, int n) {
    int i = blockIdx.x * blockDim.x + threadIdx.x;
    if (i < n) deg[i] = 1.0f;              // self-loop contributes 1
}

__global__ void k_deg(const int* __restrict__ col, float* __restrict__ deg, int E) {
    int e = blockIdx.x * blockDim.x + threadIdx.x;
    if (e < E)
        __hip_atomic_fetch_add(&deg[col[e]], 1.0f,
                               __ATOMIC_RELAXED, __HIP_MEMORY_SCOPE_AGENT);
}

__global__ void k_dinv(const float* __restrict__ deg, float* __restrict__ dinv, int n) {
    int i = blockIdx.x * blockDim.x + threadIdx.x;
    if (i < n) dinv[i] = rsqrtf(deg[i]);   // deg >= 1 always (self-loops)
}

// ---------------------------------------------------------------------------
// Y[n,128] = X[n,128] @ W[128,128]  via V_WMMA_F32_16X16X4_F32
//
// W is staged in LDS pre-swizzled into the WMMA B-operand layout:
//   entry index e = (k4*8 + t)*32 + lane  holds the float2
//   { W[k4*4+koff, t*16+m], W[k4*4+koff+1, t*16+m] },  koff=(lane>>4)*2, m=lane&15
// so the inner loop issues one aligned, bank-conflict-free ds_load_b64 per
// tile that lands directly in the even VGPR pair WMMA consumes.
// ---------------------------------------------------------------------------
__global__ __launch_bounds__(256) void k_gemm128(const float* __restrict__ X,
                                                 const float* __restrict__ W,
                                                 float* __restrict__ Y,
                                                 int nrows) {
    __shared__ float Wl[CH * CH];          // 64 KB, swizzled
    for (int e = threadIdx.x; e < CH * CH / 2; e += 256) {
        const int k4   = e >> 8;           // 0..31  (K group of 4)
        const int t    = (e >> 5) & 7;     // 0..7   (N tile)
        const int ln   = e & 31;           // lane this entry feeds
        const int koff = (ln >> 4) << 1;   // 0 | 2
        const int m    = ln & 15;
        const int k    = (k4 << 2) + koff;
        const int col  = t * 16 + m;
        Wl[e * 2 + 0] = W[(size_t)k       * CH + col];
        Wl[e * 2 + 1] = W[(size_t)(k + 1) * CH + col];
    }
    __syncthreads();

    const int wave    = threadIdx.x >> 5;
    const int lane    = threadIdx.x & 31;
    const int strip   = blockIdx.x * 8 + wave;
    const int nstrips = nrows >> 4;
    if (strip >= nstrips) return;          // wave-uniform: EXEC stays all-1s

    const int m    = lane & 15;            // row-in-tile (A) / col-in-tile (D)
    const int koff = (lane >> 4) << 1;     // 0 for lanes 0-15, 2 for lanes 16-31
    const float* xr = X + (size_t)((strip << 4) + m) * CH;

    v8f acc[8] = {};
#pragma unroll 2
    for (int k4 = 0; k4 < 32; ++k4) {
        v2f a = *(const v2f*)(xr + (k4 << 2) + koff);      // A: 16x4 f32 tile
        const float* wb = Wl + (k4 << 9) + (lane << 1);    // (k4*8+0)*64 + lane*2
#pragma unroll
        for (int t = 0; t < 8; ++t) {
            v2f b = *(const v2f*)(wb + (t << 6));          // one ds_load_b64
            acc[t] = __builtin_amdgcn_wmma_f32_16x16x4_f32(
                false, a, false, b, (short)0, acc[t], false, false);
        }
    }

    // C/D layout: VGPR r -> row r (lanes 0-15) / row r+8 (lanes 16-31), N = lane&15
    const int rbase = (strip << 4) + ((lane >> 4) << 3);
#pragma unroll
    for (int t = 0; t < 8; ++t)
#pragma unroll
        for (int r = 0; r < 8; ++r)
            Y[(size_t)(rbase + r) * CH + t * 16 + m] = acc[t][r];
}

// ---------------------------------------------------------------------------
// edge aggregation: h[col] += dinv[row]*dinv[col] * xw[row]
// one wave per edge, float4 per lane (32*4 = 128 channels)
// ---------------------------------------------------------------------------
__global__ __launch_bounds__(256) void k_agg(const float* __restrict__ xw,
                                             const int* __restrict__ ei,
                                             const float* __restrict__ dinv,
                                             float* __restrict__ h,
                                             int E) {
    const int wid = (blockIdx.x * 256 + threadIdx.x) >> 5;
    if (wid >= E) return;
    const int lane = threadIdx.x & 31;
    const int r = ei[wid];
    const int c = ei[E + wid];
    const float w = dinv[r] * dinv[c];
    const float4 v = *(const float4*)(xw + (size_t)r * CH + lane * 4);
    float* dst = h + (size_t)c * CH + lane * 4;
    __hip_atomic_fetch_add(dst + 0, w * v.x, __ATOMIC_RELAXED, __HIP_MEMORY_SCOPE_AGENT);
    __hip_atomic_fetch_add(dst + 1, w * v.y, __ATOMIC_RELAXED, __HIP_MEMORY_SCOPE_AGENT);
    __hip_atomic_fetch_add(dst + 2, w * v.z, __ATOMIC_RELAXED, __HIP_MEMORY_SCOPE_AGENT);
    __hip_atomic_fetch_add(dst + 3, w * v.w, __ATOMIC_RELAXED, __HIP_MEMORY_SCOPE_AGENT);
}

// ---------------------------------------------------------------------------
// h = (h + dinv^2 * xw + bias), optional ReLU   (self-loop term fused)
// ---------------------------------------------------------------------------
__global__ void k_post(float* __restrict__ h, const float* __restrict__ xw,
                       const float* __restrict__ dinv, const float* __restrict__ bias,
                       int n, int relu) {
    const int t = blockIdx.x * blockDim.x + threadIdx.x;
    const int node = t >> 5;
    if (node >= n) return;
    const int c4 = (t & 31) << 2;
    const float d = dinv[node];
    const float w = d * d;
    float4 hv = *(float4*)(h + (size_t)node * CH + c4);
    float4 xv = *(const float4*)(xw + (size_t)node * CH + c4);
    float4 bv = *(const float4*)(bias + c4);
    float4 o;
    o.x = hv.x + w * xv.x + bv.x;
    o.y = hv.y + w * xv.y + bv.y;
    o.z = hv.z + w * xv.z + bv.z;
    o.w = hv.w + w * xv.w + bv.w;
    if (relu) {
        o.x = fmaxf(o.x, 0.0f); o.y = fmaxf(o.y, 0.0f);
        o.z = fmaxf(o.z, 0.0f); o.w = fmaxf(o.w, 0.0f);
    }
    *(float4*)(h + (size_t)node * CH + c4) = o;
}

// ---------------------------------------------------------------------------
// segment sum pooling: one wave per node
// ---------------------------------------------------------------------------
__global__ __launch_bounds__(256) void k_pool(const float* __restrict__ h,
                                              const int* __restrict__ batch,
                                              float* __restrict__ sums,
                                              float* __restrict__ cnt,
                                              int n) {
    const int wid = (blockIdx.x * 256 + threadIdx.x) >> 5;
    if (wid >= n) return;
    const int lane = threadIdx.x & 31;
    const int g = batch[wid];
    const float4 v = *(const float4*)(h + (size_t)wid * CH + lane * 4);
    float* dst = sums + (size_t)g * CH + lane * 4;
    __hip_atomic_fetch_add(dst + 0, v.x, __ATOMIC_RELAXED, __HIP_MEMORY_SCOPE_AGENT);
    __hip_atomic_fetch_add(dst + 1, v.y, __ATOMIC_RELAXED, __HIP_MEMORY_SCOPE_AGENT);
    __hip_atomic_fetch_add(dst + 2, v.z, __ATOMIC_RELAXED, __HIP_MEMORY_SCOPE_AGENT);
    __hip_atomic_fetch_add(dst + 3, v.w, __ATOMIC_RELAXED, __HIP_MEMORY_SCOPE_AGENT);
    if (lane == 0)
        __hip_atomic_fetch_add(&cnt[g], 1.0f, __ATOMIC_RELAXED, __HIP_MEMORY_SCOPE_AGENT);
}

// ---------------------------------------------------------------------------
// out[g,j] = (sums[g,:] @ Wlin[:,j]) / max(cnt,1) + blin[j]
// ---------------------------------------------------------------------------
__global__ void k_final(const float* __restrict__ sums, const float* __restrict__ cnt,
                        const float* __restrict__ Wlin, const float* __restrict__ blin,
                        float* __restrict__ out) {
    const int t = blockIdx.x * blockDim.x + threadIdx.x;
    if (t >= N_GRAPHS * 2) return;
    const int g = t >> 1, j = t & 1;
    float s = 0.0f;
    for (int c = 0; c < CH; ++c) s += sums[(size_t)g * CH + c] * Wlin[c * 2 + j];
    out[t] = s / fmaxf(cnt[g], 1.0f) + blin[j];
}

// ---------------------------------------------------------------------------
extern "C" void kernel_launch(void* const* d_in, const int* in_sizes, int n_in,
                              void* d_out, int out_size, void* d_ws, size_t ws_size,
                              hipStream_t stream) {
    const float* x     = (const float*)d_in[0];
    const int*   ei    = (const int*)d_in[1];     // [2, E]
    const int*   batch = (const int*)d_in[2];
    const float* W1    = (const float*)d_in[3];
    const float* b1    = (const float*)d_in[4];
    const float* W2    = (const float*)d_in[5];
    const float* b2    = (const float*)d_in[6];
    const float* Wlin  = (const float*)d_in[7];
    const float* blin  = (const float*)d_in[8];
    float* out = (float*)d_out;

    const int N = in_sizes[0] / CH;
    const int E = in_sizes[1] / 2;
    const int Npad = (N + 63) & ~63;

    // workspace carve-up
    float* deg  = (float*)d_ws;
    float* dinv = deg  + Npad;
    float* bufA = dinv + Npad;                 // xw (layer1) / hw2 (layer2)
    float* bufB = bufA + (size_t)N * CH;       // h1 (layer1) / h2 (layer2)
    float* sums = bufB + (size_t)N * CH;
    float* cnt  = sums + (size_t)N_GRAPHS * CH;

    const int thr = 256;
    const dim3 blkNodes((N + thr - 1) / thr);
    const dim3 blkEdges((E + thr - 1) / thr);
    const dim3 blkGemm((N / 16 + 7) / 8);
    const dim3 blkAggE((E + 7) / 8);           // wave per edge
    const dim3 blkWaveN((N + 7) / 8);          // wave per node
    const dim3 blkPost(((size_t)N * 32 + thr - 1) / thr);

    // --- normalization -----------------------------------------------------
    k_init_deg<<<blkNodes, thr, 0, stream>>>(deg, N);
    k_deg<<<blkEdges, thr, 0, stream>>>(ei + E, deg, E);   // col = edge_index[1]
    k_dinv<<<blkNodes, thr, 0, stream>>>(deg, dinv, N);

    // --- layer 1: xw = x@W1 ; h1 = relu(scatter + selfloop + b1) -----------
    k_gemm128<<<blkGemm, thr, 0, stream>>>(x, W1, bufA, N);
    hipMemsetAsync(bufB, 0, (size_t)N * CH * sizeof(float), stream);
    k_agg<<<blkAggE, thr, 0, stream>>>(bufA, ei, dinv, bufB, E);
    k_post<<<blkPost, thr, 0, stream>>>(bufB, bufA, dinv, b1, N, 1);

    // --- layer 2: hw2 = h1@W2 ; h2 = scatter + selfloop + b2 ---------------
    k_gemm128<<<blkGemm, thr, 0, stream>>>(bufB, W2, bufA, N);
    hipMemsetAsync(bufB, 0, (size_t)N * CH * sizeof(float), stream);
    k_agg<<<blkAggE, thr, 0, stream>>>(bufA, ei, dinv, bufB, E);
    k_post<<<blkPost, thr, 0, stream>>>(bufB, bufA, dinv, b2, N, 0);

    // --- global mean pool + linear head ------------------------------------
    hipMemsetAsync(sums, 0, (size_t)N_GRAPHS * CH * sizeof(float), stream);
    hipMemsetAsync(cnt,  0, (size_t)N_GRAPHS * sizeof(float), stream);
    k_pool<<<blkWaveN, thr, 0, stream>>>(bufB, batch, sums, cnt, N);
    k_final<<<dim3((N_GRAPHS * 2 + thr - 1) / thr), thr, 0, stream>>>(
        sums, cnt, Wlin, blin, out);
}